// CustomLinear_41532333752989
// MI455X (gfx1250) — compile-verified
//
#include <hip/hip_runtime.h>
#include <math.h>

// ---------------------------------------------------------------------------
// Learned-twiddle FFT (N=8192, 13 stages) for MI455X / gfx1250, single pass.
//
// Key facts exploited:
//  * perm is j ^ 4096 + zero padding  =>  after perm, first half == 0.
//    Stages 1..12 (block size <= 4096, block-independent twiddles) keep the
//    zero half zero, so we only process the live 4096-float segment.
//  * Stage 13: out[j] = t[j], out[j+4096] = -t[j], t = tw13 * z.
//  * Stages 1..4 combined = fixed 16x16 complex matrix M applied to each
//    contiguous 16-block (input still real there) -> two real 16x16 GEMMs
//    mapped onto V_WMMA_F32_16X16X4_F32 (4 K-chunk accumulation, x2 for
//    Mr / Mi) per wave, 16 blocks per wave.
//  * Input row staged into LDS via the Tensor Data Mover (TDM) when the
//    builtin is available; otherwise plain coalesced loads.
// ---------------------------------------------------------------------------

typedef __attribute__((ext_vector_type(2)))  float        v2f;
typedef __attribute__((ext_vector_type(8)))  float        v8f;
typedef __attribute__((ext_vector_type(4)))  unsigned int u32x4;
typedef __attribute__((ext_vector_type(8)))  int          i32x8;
typedef __attribute__((ext_vector_type(4)))  int          i32x4;

#define FFT_N   8192
#define SEG     4096
#define ROWS    2048
#define WG      512
#define PI_F    3.14159265358979323846f

// ---------------------------------------------------------------------------
// Kernel 1: build twiddle heap tw[half + o] = exp(-2*pi*i/8192 * k * w[k]),
// k = o << (13 - s), for s = 1..13, plus the combined stage1-4 16x16 complex
// matrix (Mr at mats[0..255], Mi at mats[256..511]).
// ---------------------------------------------------------------------------
__global__ void fft_setup_kernel(const float* __restrict__ w,
                                 float2* __restrict__ tw,
                                 float* __restrict__ mats)
{
    const int blk = blockIdx.x;
    if (blk < 13) {
        const int s    = blk + 1;
        const int half = 1 << (s - 1);
        for (int o = threadIdx.x; o < half; o += blockDim.x) {
            const int   k   = o << (13 - s);
            const float ang = (-2.0f * PI_F / 8192.0f) * (float)k * w[k];
            float sn, cs;
            __sincosf(ang, &sn, &cs);
            tw[half + o] = make_float2(cs, sn);
        }
    } else {
        // One thread per basis column: run stages 1..4 on e_j to get M[:,j].
        const int j = threadIdx.x;
        if (j < 16) {
            float cr[16], ci[16];
            #pragma unroll
            for (int m = 0; m < 16; ++m) { cr[m] = (m == j) ? 1.0f : 0.0f; ci[m] = 0.0f; }
            for (int s = 1; s <= 4; ++s) {
                const int step = 1 << s, half = step >> 1;
                for (int b = 0; b < 16; b += step) {
                    for (int o = 0; o < half; ++o) {
                        const int   i0  = b + o, i1 = i0 + half;
                        const int   k   = o << (13 - s);
                        const float ang = (-2.0f * PI_F / 8192.0f) * (float)k * w[k];
                        float sn, cs;
                        __sincosf(ang, &sn, &cs);
                        const float tr = cs * cr[i1] - sn * ci[i1];
                        const float ti = cs * ci[i1] + sn * cr[i1];
                        cr[i1] = cr[i0] - tr;  ci[i1] = ci[i0] - ti;
                        cr[i0] = cr[i0] + tr;  ci[i0] = ci[i0] + ti;
                    }
                }
            }
            for (int m = 0; m < 16; ++m) {
                mats[      m * 16 + j] = cr[m];   // Mr row-major
                mats[256 + m * 16 + j] = ci[m];   // Mi row-major
            }
        }
    }
}

// ---------------------------------------------------------------------------
// Kernel 2: one workgroup per row. 512 threads = 16 wave32s.
// ---------------------------------------------------------------------------
__global__ __launch_bounds__(WG) void fft_main_kernel(const float* __restrict__ x,
                                                      const float2* __restrict__ tw,
                                                      const float* __restrict__ mats,
                                                      float2* __restrict__ out)
{
    __shared__ float xin[SEG];   // staged input row (real)
    __shared__ float zre[SEG];   // working real part
    __shared__ float zim[SEG];   // working imag part

    const int    row  = blockIdx.x;
    const int    tid  = threadIdx.x;
    const float* xrow = x + (size_t)row * SEG;

    // ---- stage-in: TDM DMA of the 16KB row into LDS (async-tensor path) ----
#if __has_builtin(__builtin_amdgcn_tensor_load_to_lds) && __has_builtin(__builtin_amdgcn_s_wait_tensorcnt)
    if (tid < 32) {
        if (tid == 0) {
            const unsigned           ldsoff = (unsigned)(size_t)(&xin[0]);
            const unsigned long long ga     = (unsigned long long)(size_t)xrow;
            // D# group0: count=1, lds_addr, global_addr[56:0], type=2 (bits 127:126)
            u32x4 g0 = { 1u,
                         ldsoff,
                         (unsigned)(ga & 0xFFFFFFFFu),
                         (unsigned)((ga >> 32) & 0x01FFFFFFu) | 0x80000000u };
            // D# group1: data_size=4B (code 2), tensor_dim0=4096, tensor_dim1=1,
            //            tile_dim0=4096, tile_dim1=1, dim0_stride=4096
            i32x8 g1 = { (int)(2u << 16),        // wg_mask=0, data_size=2 (4B)
                         (int)(4096u << 16),     // tensor_dim0[15:0] @ bits 63:48
                         (int)(1u << 16),        // tensor_dim0[31:16]=0, tensor_dim1[15:0]=1
                         (int)(4096u << 16),     // tensor_dim1[31:16]=0, tile_dim0=4096
                         1,                      // tile_dim1=1, tile_dim2=0
                         4096,                   // tensor_dim0_stride[31:0]
                         0,                      // stride hi + dim1_stride lo
                         0 };
            i32x4 gz  = { 0, 0, 0, 0 };
            i32x8 gz8 = { 0, 0, 0, 0, 0, 0, 0, 0 };
            __builtin_amdgcn_tensor_load_to_lds(g0, g1, gz, gz, gz8, 0);
        }
        __builtin_amdgcn_s_wait_tensorcnt(0);
    }
    __syncthreads();
#else
    for (int i = tid; i < SEG; i += WG) xin[i] = xrow[i];
    __syncthreads();
#endif

    // ---- stages 1..4 via V_WMMA_F32_16X16X4_F32: Y = M * X (real input) ----
    // Wave w handles 16 consecutive 16-blocks: elements [256w, 256w+256).
    // A = Mr/Mi K-chunk (16x4), B = 16 data blocks as columns (4x16),
    // D accumulates Y[16x16] over 4 K-chunks.
    {
        const int wave    = tid >> 5;
        const int lane    = tid & 31;
        const int ln      = lane & 15;     // column/row index within 16
        const int lh      = lane >> 4;     // half-wave select
        const int segBase = wave * 256;

        v8f yr; v8f yi;
        #pragma unroll
        for (int m = 0; m < 8; ++m) { yr[m] = 0.0f; yi[m] = 0.0f; }

        #pragma unroll
        for (int kc = 0; kc < 4; ++kc) {
            // A 16x4 layout: reg j, lane l -> M=l%16, K=j+2*(l/16)
            v2f Ar, Ai, B;
            Ar.x = mats[      ln * 16 + 4 * kc + 2 * lh + 0];
            Ar.y = mats[      ln * 16 + 4 * kc + 2 * lh + 1];
            Ai.x = mats[256 + ln * 16 + 4 * kc + 2 * lh + 0];
            Ai.y = mats[256 + ln * 16 + 4 * kc + 2 * lh + 1];
            // B 4x16: reg j, lane l -> N=l%16 (block), K=j+2*(l/16)
            B.x  = xin[segBase + 16 * ln + 4 * kc + 2 * lh + 0];
            B.y  = xin[segBase + 16 * ln + 4 * kc + 2 * lh + 1];

            yr = __builtin_amdgcn_wmma_f32_16x16x4_f32(false, Ar, false, B,
                                                       (short)0, yr, false, false);
            yi = __builtin_amdgcn_wmma_f32_16x16x4_f32(false, Ai, false, B,
                                                       (short)0, yi, false, false);
        }

        // D layout: VGPR m, lanes 0-15 -> M=m, lanes 16-31 -> M=m+8; N = l%16.
        // Element index = segBase + 16*N + M  -> 8 consecutive floats per lane.
        const int dbase = segBase + 16 * ln + 8 * lh;
        #pragma unroll
        for (int m = 0; m < 8; ++m) {
            zre[dbase + m] = yr[m];
            zim[dbase + m] = yi[m];
        }
    }
    __syncthreads();

    // ---- stages 5..12: classic butterflies in LDS (2048 per stage) ----
    #pragma unroll 1
    for (int s = 5; s <= 12; ++s) {
        const int half = 1 << (s - 1);
        for (int b = tid; b < SEG / 2; b += WG) {
            const int    o  = b & (half - 1);
            const int    i0 = ((b >> (s - 1)) << s) + o;
            const int    i1 = i0 + half;
            const float2 t  = tw[half + o];
            const float  hr = zre[i1], hi = zim[i1];
            const float  trv = t.x * hr - t.y * hi;
            const float  tiv = t.x * hi + t.y * hr;
            const float  lr = zre[i0], li = zim[i0];
            zre[i1] = lr - trv;  zim[i1] = li - tiv;
            zre[i0] = lr + trv;  zim[i0] = li + tiv;
        }
        __syncthreads();
    }

    // ---- stage 13 fused with store: lo half is zero -> out = [t, -t] ----
    float2* orow = out + (size_t)row * FFT_N;
    for (int j = tid; j < SEG; j += WG) {
        const float2 t  = tw[4096 + j];
        const float  zr = zre[j], zi = zim[j];
        const float  trv = t.x * zr - t.y * zi;
        const float  tiv = t.x * zi + t.y * zr;
        orow[j]        = make_float2( trv,  tiv);
        orow[j + SEG]  = make_float2(-trv, -tiv);
    }
}

// ---------------------------------------------------------------------------
extern "C" void kernel_launch(void* const* d_in, const int* in_sizes, int n_in,
                              void* d_out, int out_size, void* d_ws, size_t ws_size,
                              hipStream_t stream)
{
    const float* x = (const float*)d_in[0];   // [2048, 4096] f32
    const float* w = (const float*)d_in[1];   // [8192] f32

    float2* tw   = (float2*)d_ws;                                  // 8192 float2 (heap)
    float*  mats = (float*)((char*)d_ws + FFT_N * sizeof(float2)); // 512 f32 (Mr|Mi)
    float2* out  = (float2*)d_out;                                 // [2048, 8192] complex64

    fft_setup_kernel<<<14, 256, 0, stream>>>(w, tw, mats);
    fft_main_kernel<<<ROWS, WG, 0, stream>>>(x, tw, mats, out);
}